// CausalConvolution_88862873354963
// MI455X (gfx1250) — compile-verified
//
#include <hip/hip_runtime.h>
#include <math.h>

// Problem constants (from reference): B=4, S=4096, C=2048, K=4, fp32.
namespace {
constexpr int B_ = 4;
constexpr int S_ = 4096;
constexpr int C_ = 2048;
constexpr int K_ = 4;

constexpr int TS      = 32;              // output timesteps per tile
constexpr int ROWS    = TS + K_ - 1;     // 35 input rows (incl. 3-row halo)
constexpr int THREADS = 256;             // threads per workgroup (8 waves, wave32)
constexpr int CHUNK   = THREADS * 4;     // 1024 channels per workgroup (float4/thread)
constexpr size_t LDS_BYTES = (size_t)ROWS * THREADS * sizeof(float4); // 140 KB
}

typedef float f32x4 __attribute__((ext_vector_type(4)));  // native vector for NT store

// y[b,s,c] = mask[b,s] * silu( b[c] + sum_k W[c,k] * x[b, s-3+k, c] * mask[b, s-3+k] )
__global__ __launch_bounds__(THREADS) void causal_conv_silu_kernel(
    const float* __restrict__ hs, const float* __restrict__ mask,
    const float* __restrict__ W, const float* __restrict__ bias,
    float* __restrict__ y)
{
  extern __shared__ float4 tile[];                 // [ROWS][THREADS]
  const int t  = threadIdx.x;
  const int s0 = blockIdx.y * TS;                  // tile start along S
  const int b  = blockIdx.z;
  const int c0 = blockIdx.x * CHUNK + t * 4;       // first of this thread's 4 channels

  const float* gbase = hs + (size_t)b * S_ * C_ + c0;

  // ---- Stage tile into LDS with CDNA5 async global->LDS copies (ASYNCcnt) ----
  #pragma unroll
  for (int r = 0; r < ROWS; ++r) {
    const int sp = s0 - (K_ - 1) + r;              // global timestep of row r (uniform)
    float4* lp = &tile[r * THREADS + t];
    if (sp >= 0) {
      const float* gaddr = gbase + (size_t)sp * C_;
      unsigned lds_off = (unsigned)(size_t)(void*)lp;  // low 32b of flat ptr = LDS offset
      asm volatile("global_load_async_to_lds_b128 %0, %1, off"
                   :: "v"(lds_off), "v"(gaddr)
                   : "memory");
    } else {
      *lp = make_float4(0.f, 0.f, 0.f, 0.f);       // causal zero padding (first tile only)
    }
  }
  asm volatile("s_wait_asynccnt 0x0" ::: "memory");
  __syncthreads();

  // ---- Per-thread constants: 4 channels x 4 taps, bias ----
  const float4 w0 = *(const float4*)(W + (size_t)c0 * K_ + 0);
  const float4 w1 = *(const float4*)(W + (size_t)c0 * K_ + 4);
  const float4 w2 = *(const float4*)(W + (size_t)c0 * K_ + 8);
  const float4 w3 = *(const float4*)(W + (size_t)c0 * K_ + 12);
  const float4 bb = *(const float4*)(bias + c0);
  const float* mrow = mask + (size_t)b * S_;

  auto loadrow = [&](int r) -> float4 {
    const int sp = s0 - (K_ - 1) + r;
    const float m = (sp >= 0) ? mrow[sp] : 0.0f;   // input mask premultiply
    float4 v = tile[r * THREADS + t];
    v.x *= m; v.y *= m; v.z *= m; v.w *= m;
    return v;
  };

  // ---- Sliding 4-row window over the tile ----
  float4 x0 = loadrow(0);
  float4 x1 = loadrow(1);
  float4 x2 = loadrow(2);

  float* ybase = y + ((size_t)b * S_ + s0) * C_ + c0;

  #pragma unroll 4
  for (int i = 0; i < TS; ++i) {
    const float4 x3 = loadrow(i + K_ - 1);
    float4 a;
    a.x = fmaf(w0.x, x0.x, fmaf(w0.y, x1.x, fmaf(w0.z, x2.x, fmaf(w0.w, x3.x, bb.x))));
    a.y = fmaf(w1.x, x0.y, fmaf(w1.y, x1.y, fmaf(w1.z, x2.y, fmaf(w1.w, x3.y, bb.y))));
    a.z = fmaf(w2.x, x0.z, fmaf(w2.y, x1.z, fmaf(w2.z, x2.z, fmaf(w2.w, x3.z, bb.z))));
    a.w = fmaf(w3.x, x0.w, fmaf(w3.y, x1.w, fmaf(w3.z, x2.w, fmaf(w3.w, x3.w, bb.w))));

    const float mo = mrow[s0 + i];                 // output mask
    f32x4 av;
    av.x = mo * (a.x / (1.0f + expf(-a.x)));       // silu
    av.y = mo * (a.y / (1.0f + expf(-a.y)));
    av.z = mo * (a.z / (1.0f + expf(-a.z)));
    av.w = mo * (a.w / (1.0f + expf(-a.w)));

    f32x4* dst = (f32x4*)(ybase + (size_t)i * C_);
    __builtin_nontemporal_store(av, dst);          // streaming output, keep L2 for x

    x0 = x1; x1 = x2; x2 = x3;
  }
}

// input_state[b,c,k] = hs[b, S-K+k, c] * mask[b, S-K+k]
__global__ __launch_bounds__(256) void conv_state_kernel(
    const float* __restrict__ hs, const float* __restrict__ mask,
    float* __restrict__ st)
{
  const int idx = blockIdx.x * blockDim.x + threadIdx.x;   // over B*C
  if (idx >= B_ * C_) return;
  const int b = idx / C_;
  const int c = idx - b * C_;
  float4 v;
  float* vp = (float*)&v;
  #pragma unroll
  for (int k = 0; k < K_; ++k) {
    const int sp = S_ - K_ + k;
    vp[k] = hs[((size_t)b * S_ + sp) * C_ + c] * mask[(size_t)b * S_ + sp];
  }
  *(float4*)(st + ((size_t)b * C_ + c) * K_) = v;
}

extern "C" void kernel_launch(void* const* d_in, const int* in_sizes, int n_in,
                              void* d_out, int out_size, void* d_ws, size_t ws_size,
                              hipStream_t stream) {
  (void)in_sizes; (void)n_in; (void)out_size; (void)d_ws; (void)ws_size;
  const float* hs   = (const float*)d_in[0];   // (B,S,C) fp32
  const float* mask = (const float*)d_in[1];   // (B,S)   fp32
  const float* W    = (const float*)d_in[2];   // (C,1,K) fp32
  const float* bias = (const float*)d_in[3];   // (C,)    fp32

  float* y  = (float*)d_out;                           // (B,S,C)
  float* st = y + (size_t)B_ * S_ * C_;                // (B,C,K)

  dim3 grid(C_ / CHUNK, S_ / TS, B_);                  // (2, 128, 4)
  causal_conv_silu_kernel<<<grid, THREADS, LDS_BYTES, stream>>>(hs, mask, W, bias, y);

  conv_state_kernel<<<(B_ * C_ + 255) / 256, 256, 0, stream>>>(hs, mask, st);
}